// MultiModelLinear_14723147890775
// MI455X (gfx1250) — compile-verified
//
#include <hip/hip_runtime.h>

typedef __attribute__((ext_vector_type(2))) float v2f;
typedef __attribute__((ext_vector_type(8))) float v8f;

#define NUM_MODELS 64
#define FEAT 128

// ---------------- Pass 1: counting sort of sample ids by model index ----------------

__global__ void mml_zero(int* counts, int* cursors) {
    int t = threadIdx.x;
    if (t < NUM_MODELS) { counts[t] = 0; cursors[t] = 0; }
}

__global__ void mml_count(const int* __restrict__ idx, int* __restrict__ counts, int n) {
    int i = blockIdx.x * blockDim.x + threadIdx.x;
    if (i < n) atomicAdd(&counts[idx[i]], 1);
}

__global__ void mml_scan(const int* __restrict__ counts, int* __restrict__ offsets) {
    if (threadIdx.x == 0) {
        int acc = 0;
        for (int m = 0; m < NUM_MODELS; ++m) { offsets[m] = acc; acc += counts[m]; }
        offsets[NUM_MODELS] = acc;
    }
}

__global__ void mml_scatter(const int* __restrict__ idx, const int* __restrict__ offsets,
                            int* __restrict__ cursors, int* __restrict__ perm, int n) {
    int i = blockIdx.x * blockDim.x + threadIdx.x;
    if (i < n) {
        int m = idx[i];
        int pos = atomicAdd(&cursors[m], 1);
        perm[offsets[m] + pos] = i;
    }
}

// ---------------- Pass 2: grouped GEMM via V_WMMA_F32_16X16X4_F32 ----------------
//
// Block: 256 threads = 8 waves; blockIdx.x = model, blockIdx.y strides 16-row chunks.
// Wave w owns output columns [16w, 16w+16). K = 128 -> 32 WMMA steps of K=4.
//
// fp32 WMMA fragment layouts (ISA 7.12.2):
//   A 16x4 : lane L holds row M=L&15; VGPR j holds K = 4*kb + 2*(L>>4) + j
//   B 4x16 : lane L holds col N=L&15; VGPR j holds K = 4*kb + 2*(L>>4) + j
//   C 16x16: lane L, VGPR r -> row M = r + 8*(L>>4), col N = L&15

__global__ void __launch_bounds__(256) mml_wmma_gemm(
        const float* __restrict__ X, const float* __restrict__ W,
        const float* __restrict__ Bias, const int* __restrict__ perm,
        const int* __restrict__ offsets, float* __restrict__ Out) {

    const int m     = blockIdx.x;
    const int begin = offsets[m];
    const int end   = offsets[m + 1];
    if (begin >= end) return;                 // uniform across block

    const int lane = threadIdx.x & 31;
    const int wave = threadIdx.x >> 5;        // 0..7 -> column tile
    const int half = lane >> 4;               // 0 or 1
    const int l16  = lane & 15;
    const int col  = wave * 16 + l16;         // global output column

    // Preload B fragments: weights[m, col, k] for all k (read once per block).
    const v2f* wrow = reinterpret_cast<const v2f*>(W + ((size_t)m * FEAT + col) * FEAT);
    v2f bfrag[32];
    #pragma unroll
    for (int kb = 0; kb < 32; ++kb)
        bfrag[kb] = wrow[2 * kb + half];      // elems 4*kb + 2*half, +1

    const float biasv = Bias[m * FEAT + col];

    __shared__ int s_sid[16];

    for (int base = begin + blockIdx.y * 16; base < end; base += gridDim.y * 16) {
        __syncthreads();                      // protect s_sid WAR across iterations
        if (threadIdx.x < 16) {
            int r = base + threadIdx.x;
            s_sid[threadIdx.x] = perm[r < end ? r : begin];  // pad tail with a valid row
        }
        __syncthreads();

        const int sid_a = s_sid[l16];
        const v2f* xrow = reinterpret_cast<const v2f*>(X + (size_t)sid_a * FEAT);

        v8f acc = {};
        #pragma unroll
        for (int kb = 0; kb < 32; ++kb) {
            v2f a = xrow[2 * kb + half];
            acc = __builtin_amdgcn_wmma_f32_16x16x4_f32(
                /*neg_a=*/false, a, /*neg_b=*/false, bfrag[kb],
                /*c_mod=*/(short)0, acc, /*reuse_a=*/false, /*reuse_b=*/false);
        }

        #pragma unroll
        for (int r = 0; r < 8; ++r) {
            int M  = r + 8 * half;
            int gr = base + M;
            if (gr < end) {
                int sid = s_sid[M];
                Out[(size_t)sid * FEAT + col] = acc[r] + biasv;
            }
        }
    }
}

// ---------------- Launch ----------------

extern "C" void kernel_launch(void* const* d_in, const int* in_sizes, int n_in,
                              void* d_out, int out_size, void* d_ws, size_t ws_size,
                              hipStream_t stream) {
    const float* X   = (const float*)d_in[0];   // [N,128]
    const int*   idx = (const int*)  d_in[1];   // [N]
    const float* W   = (const float*)d_in[2];   // [64,128,128]
    const float* B   = (const float*)d_in[3];   // [64,128]
    float*       Out = (float*)d_out;           // [N,128]
    const int n = in_sizes[1];

    int* counts  = (int*)d_ws;        // 64
    int* cursors = counts + 64;       // 64
    int* offsets = cursors + 64;      // 65 (padded to 128)
    int* perm    = offsets + 128;     // n ints

    mml_zero<<<1, 64, 0, stream>>>(counts, cursors);
    mml_count<<<(n + 255) / 256, 256, 0, stream>>>(idx, counts, n);
    mml_scan<<<1, 32, 0, stream>>>(counts, offsets);
    mml_scatter<<<(n + 255) / 256, 256, 0, stream>>>(idx, offsets, cursors, perm, n);

    dim3 grid(NUM_MODELS, 8);
    mml_wmma_gemm<<<grid, 256, 0, stream>>>(X, W, B, perm, offsets, Out);
}